// Model_18159121727917
// MI455X (gfx1250) — compile-verified
//
#include <hip/hip_runtime.h>
#include <math.h>

typedef __attribute__((ext_vector_type(2))) float v2f;
typedef __attribute__((ext_vector_type(8))) float v8f;

#define B_      64
#define L_      2048
#define C_      21
#define D_      128
#define PL_     16
#define NWIN    255
#define N0      213          // first window index that touches the PRED tail
#define N_USED  42           // windows 213..254
#define PRED_   336
#define T0      1712         // TOTAL - PRED
#define PIF     3.14159265358979323846f

// d_ws layout (float offsets)
#define WS_MEAN  0                     // 1344
#define WS_SCALE 1344                  // 1344
#define WS_WEFF  2688                  // 128*16 = 2048
#define WS_BEFF  4736                  // 16
#define WS_P     4752                  // 64*42*21*16 = 903168

// ---------------- Kernel 1: per-(b,c) mean & sqrt(var+eps) -----------------
__global__ __launch_bounds__(352) void stats_kernel(const float* __restrict__ x,
                                                    float* __restrict__ ws) {
    __shared__ float s_sum[C_];
    __shared__ float s_sq[C_];
    const int b   = blockIdx.x;
    const int tid = threadIdx.x;
    if (tid < C_) { s_sum[tid] = 0.0f; s_sq[tid] = 0.0f; }
    __syncthreads();

    if (tid < C_ * 16) {                 // 336 workers
        const int c  = tid % C_;
        const int lg = tid / C_;         // 0..15, each covers 128 l values
        const float* xb = x + (size_t)b * L_ * C_;
        float ls = 0.0f, lq = 0.0f;
        const int lbase = lg * 128;
        for (int i = 0; i < 128; ++i) {
            float v = xb[(size_t)(lbase + i) * C_ + c];
            ls += v;
            lq += v * v;
        }
        atomicAdd(&s_sum[c], ls);
        atomicAdd(&s_sq[c],  lq);
    }
    __syncthreads();

    if (tid < C_) {
        float s = s_sum[tid], q = s_sq[tid];
        float mean = s / (float)L_;
        float var  = (q - s * s / (float)L_) / (float)(L_ - 1) + 1e-5f;
        ws[WS_MEAN  + b * C_ + tid] = mean;
        ws[WS_SCALE + b * C_ + tid] = sqrtf(var);
    }
}

// ------------- Kernel 2: fuse W_head (D x PL) with IDCT: W_eff, b_eff ------
__global__ __launch_bounds__(256) void weights_kernel(const float* __restrict__ Wh,
                                                      const float* __restrict__ bh,
                                                      float* __restrict__ ws) {
    const int e = blockIdx.x * blockDim.x + threadIdx.x;
    if (e >= D_ * PL_ + PL_) return;
    if (e < D_ * PL_) {
        const int d = e >> 4;
        const int m = e & 15;
        float acc = 0.0f;
        for (int k = 0; k < PL_; ++k) {
            float w = (k == 0) ? (1.0f / 32.0f) : (1.0f / 16.0f);
            acc += Wh[d * PL_ + k] * cosf(PIF * (2.0f * m + 1.0f) * k / 32.0f) * w;
        }
        ws[WS_WEFF + e] = acc;
    } else {
        const int m = e - D_ * PL_;
        float acc = 0.0f;
        for (int k = 0; k < PL_; ++k) {
            float w = (k == 0) ? (1.0f / 32.0f) : (1.0f / 16.0f);
            acc += bh[k] * cosf(PIF * (2.0f * m + 1.0f) * k / 32.0f) * w;
        }
        ws[WS_BEFF + m] = acc;
    }
}

// ------------- Kernel 3: WMMA GEMM  p = z[:,213:255] @ W_eff + b_eff -------
// One wave per 16x16 output tile. Rows indexed (b, ni, c) -> gid.
// 3528 tiles total, 8 waves/block -> 441 blocks.
__global__ __launch_bounds__(256) void gemm_kernel(const float* __restrict__ z,
                                                   float* __restrict__ ws) {
    __shared__ float sW[D_ * PL_];
    __shared__ float sb[PL_];
    const int tid = threadIdx.x;
    for (int i = tid; i < D_ * PL_; i += 256) sW[i] = ws[WS_WEFF + i];
    if (tid < PL_) sb[tid] = ws[WS_BEFF + tid];
    __syncthreads();

    const int wave = tid >> 5;
    const int lane = tid & 31;
    const int tile = blockIdx.x * 8 + wave;
    const int r0   = tile * 16;

    const int m    = lane & 15;          // A-row / B-col / C-col for this lane
    const int koff = (lane >> 4) << 1;   // 0 or 2: which K pair this lane holds

    // map A-row gid -> (b, ni, c) -> z row pointer
    const int gid = r0 + m;
    const int c   = gid % C_;
    const int t1  = gid / C_;
    const int ni  = t1 % N_USED;
    const int b   = t1 / N_USED;
    const float* zrow = z + ((size_t)(b * NWIN + (N0 + ni)) * C_ + c) * D_;

    v8f acc = {};
#pragma unroll
    for (int kb = 0; kb < D_; kb += 4) {
        // A fragment: A[m][kb+koff], A[m][kb+koff+1]
        v2f a = *(const v2f*)(zrow + kb + koff);
        // B fragment: B[kb+koff][n], B[kb+koff+1][n]  (n == lane&15)
        v2f bb;
        const int bi = (kb + koff) * PL_ + m;
        bb.x = sW[bi];
        bb.y = sW[bi + PL_];
        acc = __builtin_amdgcn_wmma_f32_16x16x4_f32(
            /*neg_a=*/false, a, /*neg_b=*/false, bb,
            /*c_mod=*/(short)0, acc, /*reuse_a=*/false, /*reuse_b=*/false);
    }

    const float bias = sb[m];
    float* p = ws + WS_P;
    const int half = (lane >> 4) << 3;   // rows 0..7 or 8..15
#pragma unroll
    for (int r = 0; r < 8; ++r) {
        const int grow = r0 + r + half;
        p[(size_t)grow * PL_ + m] = acc[r] + bias;
    }
}

// ------------- Kernel 4: overlap-add tail + affine de-normalization --------
__global__ __launch_bounds__(256) void out_kernel(const float* __restrict__ ws,
                                                  float* __restrict__ out) {
    const int idx = blockIdx.x * blockDim.x + threadIdx.x;
    if (idx >= B_ * C_ * PRED_) return;
    const int j  = idx % PRED_;
    const int bc = idx / PRED_;
    const int c  = bc % C_;
    const int b  = bc / C_;
    const int t  = T0 + j;
    const int nf = t >> 3;       // candidate window floor(t/8)
    const int l2 = t & 7;

    const float* p = ws + WS_P;
    const int n1 = nf - 1;       // always valid for t >= 1712 (n1 >= 213)
    float sum = p[(size_t)(((b * N_USED) + (n1 - N0)) * C_ + c) * PL_ + (l2 + 8)];
    float inv = 1.0f;
    if (nf <= NWIN - 1) {
        sum += p[(size_t)(((b * N_USED) + (nf - N0)) * C_ + c) * PL_ + l2];
        inv = 0.5f;
    }
    out[idx] = (sum * inv) * ws[WS_SCALE + bc] + ws[WS_MEAN + bc];
}

extern "C" void kernel_launch(void* const* d_in, const int* in_sizes, int n_in,
                              void* d_out, int out_size, void* d_ws, size_t ws_size,
                              hipStream_t stream) {
    const float* x  = (const float*)d_in[0];
    const float* z  = (const float*)d_in[1];
    const float* Wh = (const float*)d_in[2];
    const float* bh = (const float*)d_in[3];
    float* ws  = (float*)d_ws;
    float* out = (float*)d_out;

    stats_kernel<<<B_, 352, 0, stream>>>(x, ws);
    weights_kernel<<<(D_ * PL_ + PL_ + 255) / 256, 256, 0, stream>>>(Wh, bh, ws);
    gemm_kernel<<<441, 256, 0, stream>>>(z, ws);           // 3528 tiles / 8 waves
    const int total = B_ * C_ * PRED_;
    out_kernel<<<(total + 255) / 256, 256, 0, stream>>>(ws, out);
}